// TimeSeriesPixelEncoder_84628035601132
// MI455X (gfx1250) — compile-verified
//
#include <hip/hip_runtime.h>
#include <hip/hip_bf16.h>
#include <math.h>

// Problem constants (from reference): x [16, 1024, 8] f32 -> out [16, 3, 224, 224] f32
#define B_    16
#define L_    1024
#define D_    8
#define BD_   (B_*D_)
#define IMG_  224
#define EPS_  1e-8f
#define SPLIT_ 8               // blocks per (b,d) in the WMMA min/max sweep
#define FLT_BIG 3.402823466e+38f

typedef __attribute__((ext_vector_type(2))) float v2f;
typedef __attribute__((ext_vector_type(8))) float v8f;

// ---------------- stage 1: global min/max of x ----------------
__global__ void __launch_bounds__(256) k_xminmax_s1(const float* __restrict__ x, int n,
                                                    float* __restrict__ part) {
    __shared__ float rlo[256], rhi[256];
    int tid = threadIdx.x;
    float lo = FLT_BIG, hi = -FLT_BIG;
    for (int i = blockIdx.x * blockDim.x + tid; i < n; i += gridDim.x * blockDim.x) {
        float v = x[i];
        lo = fminf(lo, v); hi = fmaxf(hi, v);
    }
    rlo[tid] = lo; rhi[tid] = hi; __syncthreads();
    for (int off = 128; off > 0; off >>= 1) {
        if (tid < off) {
            rlo[tid] = fminf(rlo[tid], rlo[tid + off]);
            rhi[tid] = fmaxf(rhi[tid], rhi[tid + off]);
        }
        __syncthreads();
    }
    if (tid == 0) { part[2 * blockIdx.x] = rlo[0]; part[2 * blockIdx.x + 1] = rhi[0]; }
}

// generic finalize: reduce npart (lo,hi) pairs -> dst[0]=lo, dst[1]=hi
__global__ void __launch_bounds__(256) k_minmax_fin(const float* __restrict__ part, int npart,
                                                    float* __restrict__ dst) {
    __shared__ float rlo[256], rhi[256];
    int tid = threadIdx.x;
    float lo = FLT_BIG, hi = -FLT_BIG;
    for (int i = tid; i < npart; i += 256) {
        lo = fminf(lo, part[2 * i]);
        hi = fmaxf(hi, part[2 * i + 1]);
    }
    rlo[tid] = lo; rhi[tid] = hi; __syncthreads();
    for (int off = 128; off > 0; off >>= 1) {
        if (tid < off) {
            rlo[tid] = fminf(rlo[tid], rlo[tid + off]);
            rhi[tid] = fmaxf(rhi[tid], rhi[tid + off]);
        }
        __syncthreads();
    }
    if (tid == 0) { dst[0] = rlo[0]; dst[1] = rhi[0]; }
}

// ---------------- stage 2: c = cos(theta) = clipped x_norm, s = sqrt(1-c^2) ----------------
// layout: c[(b*D+d)*L + i]  (coalesced writes; x read is stride-D but L2-resident)
__global__ void __launch_bounds__(256) k_compute_cs(const float* __restrict__ x,
                                                    const float* __restrict__ xmm,
                                                    float* __restrict__ cArr,
                                                    float* __restrict__ sArr) {
    int tid = blockIdx.x * blockDim.x + threadIdx.x;   // 0 .. BD_*L_-1
    int bd = tid / L_;
    int i  = tid % L_;
    int b  = bd / D_;
    int d  = bd % D_;
    float lo = xmm[0], hi = xmm[1];
    float rng = hi - lo;
    float v  = x[b * (L_ * D_) + i * D_ + d];
    float xn = (rng < EPS_) ? 0.0f : (v - lo) / (rng + EPS_);
    float c  = 2.0f * xn - 1.0f;
    c = fminf(fmaxf(c, -1.0f + 1e-6f), 1.0f - 1e-6f);
    cArr[tid] = c;
    sArr[tid] = sqrtf(fmaxf(0.0f, 1.0f - c * c));      // theta in (0,pi) -> sin >= 0
}

// ---------------- stage 3: per-(b,d) min/max of M[i,j] = c_i c_j - s_i s_j via WMMA ----------------
// grid = BD_*SPLIT_ blocks of 256 (8 waves). Wave-slot `slot` owns column strip tj = slot,
// ti = 0..63 (B fragment is loop-invariant). Lanes 16-31 (K=2,3 zero pad) read from a
// zeroed LDS pad region via a branchless address select -> single unconditional ds_load.
__global__ void __launch_bounds__(256) k_tile_minmax(const float* __restrict__ cArr,
                                                     const float* __restrict__ sArr,
                                                     float* __restrict__ bdpart) {
    __shared__ float cS[L_ + 16];
    __shared__ float sS[L_ + 16];
    __shared__ float rlo[256], rhi[256];
    int tid = threadIdx.x;
    int bd  = blockIdx.x / SPLIT_;
    int prt = blockIdx.x % SPLIT_;
    for (int k = tid; k < L_; k += 256) {
        cS[k] = cArr[bd * L_ + k];
        sS[k] = sArr[bd * L_ + k];
    }
    if (tid < 16) { cS[L_ + tid] = 0.0f; sS[L_ + tid] = 0.0f; }  // zero pad for K=2,3
    __syncthreads();

    int wave = tid >> 5;
    int lane = tid & 31;
    int slot = prt * 8 + wave;            // 0..63 wave-slots per (b,d) == tj strip
    int m    = lane & 15;
    bool lo16 = (lane < 16);

    // B fragment (4x16): row K=0 -> c_j, row K=1 -> -s_j; K=2,3 read zeros from pad.
    int bidx = (lo16 ? slot * 16 : L_) + m;
    v2f Bm;
    Bm.x =  cS[bidx];
    Bm.y = -sS[bidx];

    float vmin = FLT_BIG, vmax = -FLT_BIG;
#pragma unroll 4
    for (int ti = 0; ti < 64; ti++) {
        int aidx = (lo16 ? ti * 16 : L_) + m;
        v2f A;
        A.x = cS[aidx];                   // K=0 (lanes 16-31: zero pad = K=2)
        A.y = sS[aidx];                   // K=1 (lanes 16-31: zero pad = K=3)
        v8f acc = {};
        acc = __builtin_amdgcn_wmma_f32_16x16x4_f32(false, A, false, Bm,
                                                    (short)0, acc, false, false);
#pragma unroll
        for (int k = 0; k < 8; k++) {
            vmin = fminf(vmin, acc[k]);
            vmax = fmaxf(vmax, acc[k]);
        }
    }
    rlo[tid] = vmin; rhi[tid] = vmax; __syncthreads();
    for (int off = 128; off > 0; off >>= 1) {
        if (tid < off) {
            rlo[tid] = fminf(rlo[tid], rlo[tid + off]);
            rhi[tid] = fmaxf(rhi[tid], rhi[tid + off]);
        }
        __syncthreads();
    }
    if (tid == 0) { bdpart[2 * blockIdx.x] = rlo[0]; bdpart[2 * blockIdx.x + 1] = rhi[0]; }
}

__global__ void k_bd_fin(const float* __restrict__ bdpart, float* __restrict__ bdmm) {
    int bd = threadIdx.x;
    if (bd >= BD_) return;
    float lo = FLT_BIG, hi = -FLT_BIG;
    for (int p = 0; p < SPLIT_; p++) {
        lo = fminf(lo, bdpart[2 * (bd * SPLIT_ + p)]);
        hi = fmaxf(hi, bdpart[2 * (bd * SPLIT_ + p) + 1]);
    }
    bdmm[2 * bd] = lo; bdmm[2 * bd + 1] = hi;
}

// ---------------- stage 4: fused normalize + d-mean + bilinear resize (half-pixel) ----------------
__global__ void __launch_bounds__(256) k_resize(const float* __restrict__ cArr,
                                                const float* __restrict__ sArr,
                                                const float* __restrict__ bdmm,
                                                float* __restrict__ rArr,
                                                float* __restrict__ part) {
    __shared__ float rlo[256], rhi[256];
    int tid = threadIdx.x;
    int g = blockIdx.x * 256 + tid;                 // 0 .. B_*IMG_*IMG_-1 (exact)
    int b = g / (IMG_ * IMG_);
    int p = g % (IMG_ * IMG_);
    int oy = p / IMG_, ox = p % IMG_;
    const float scale = (float)L_ / (float)IMG_;
    float yin = (oy + 0.5f) * scale - 0.5f;
    float xin = (ox + 0.5f) * scale - 0.5f;
    int y0 = (int)floorf(yin); float fy = yin - (float)y0;
    int x0 = (int)floorf(xin); float fx = xin - (float)x0;
    int y0c = min(max(y0, 0), L_ - 1), y1c = min(max(y0 + 1, 0), L_ - 1);
    int x0c = min(max(x0, 0), L_ - 1), x1c = min(max(x0 + 1, 0), L_ - 1);

    float acc = 0.0f;
#pragma unroll
    for (int d = 0; d < D_; d++) {
        int base = (b * D_ + d) * L_;
        float cy0 = cArr[base + y0c], cy1 = cArr[base + y1c];
        float sy0 = sArr[base + y0c], sy1 = sArr[base + y1c];
        float cx0 = cArr[base + x0c], cx1 = cArr[base + x1c];
        float sx0 = sArr[base + x0c], sx1 = sArr[base + x1c];
        float m00 = cy0 * cx0 - sy0 * sx0;
        float m01 = cy0 * cx1 - sy0 * sx1;
        float m10 = cy1 * cx0 - sy1 * sx0;
        float m11 = cy1 * cx1 - sy1 * sx1;
        float mtop = m00 + fx * (m01 - m00);
        float mbot = m10 + fx * (m11 - m10);
        float mval = mtop + fy * (mbot - mtop);
        // normalization is affine -> commutes with bilinear interp
        float lo = bdmm[2 * (b * D_ + d)], hi = bdmm[2 * (b * D_ + d) + 1];
        float rng = hi - lo;
        acc += (rng < EPS_) ? 0.0f : (mval - lo) / (rng + EPS_);
    }
    float r = acc * (1.0f / (float)D_);
    rArr[g] = r;

    rlo[tid] = r; rhi[tid] = r; __syncthreads();
    for (int off = 128; off > 0; off >>= 1) {
        if (tid < off) {
            rlo[tid] = fminf(rlo[tid], rlo[tid + off]);
            rhi[tid] = fmaxf(rhi[tid], rhi[tid + off]);
        }
        __syncthreads();
    }
    if (tid == 0) { part[2 * blockIdx.x] = rlo[0]; part[2 * blockIdx.x + 1] = rhi[0]; }
}

// ---------------- stage 5: final global shift/scale, replicate to 3 channels ----------------
__global__ void __launch_bounds__(256) k_writeout(const float* __restrict__ rArr,
                                                  const float* __restrict__ fmm,
                                                  float* __restrict__ out, int n) {
    int g = blockIdx.x * 256 + threadIdx.x;
    if (g >= n) return;
    const int per_b = 3 * IMG_ * IMG_;
    int b = g / per_b;
    int p = (g % per_b) % (IMG_ * IMG_);
    float mn = fmm[0], mx = fmm[1];
    float v = rArr[b * (IMG_ * IMG_) + p];
    // out = (gaf - min); out /= (out.max() + 1e-6)  ->  (v - mn) / ((mx - mn) + 1e-6)
    out[g] = (v - mn) / ((mx - mn) + 1e-6f);
}

extern "C" void kernel_launch(void* const* d_in, const int* in_sizes, int n_in,
                              void* d_out, int out_size, void* d_ws, size_t ws_size,
                              hipStream_t stream) {
    (void)n_in; (void)out_size; (void)ws_size;
    const float* x = (const float*)d_in[0];
    float* out = (float*)d_out;
    const int nx = in_sizes[0];                 // 16*1024*8 = 131072

    // workspace layout (floats), total ~4.3 MB
    float* ws     = (float*)d_ws;
    float* cArr   = ws;                         // BD_*L_           = 131072
    float* sArr   = cArr + BD_ * L_;            // BD_*L_           = 131072
    float* xmm    = sArr + BD_ * L_;            // 2
    float* part   = xmm + 2;                    // up to 3136 pairs (reserve 8192 floats)
    float* bdpart = part + 8192;                // 2*BD_*SPLIT_     = 2048
    float* bdmm   = bdpart + 2 * BD_ * SPLIT_;  // 2*BD_            = 256
    float* rArr   = bdmm + 2 * BD_;             // B_*IMG_*IMG_     = 802816
    float* fmm    = rArr + B_ * IMG_ * IMG_;    // 2

    const int NPIX = B_ * IMG_ * IMG_;          // 802816 = 3136 * 256
    const int NOUT = B_ * 3 * IMG_ * IMG_;      // 2408448 = 9408 * 256

    k_xminmax_s1 <<<256, 256, 0, stream>>>(x, nx, part);
    k_minmax_fin <<<1, 256, 0, stream>>>(part, 256, xmm);
    k_compute_cs <<<(BD_ * L_) / 256, 256, 0, stream>>>(x, xmm, cArr, sArr);
    k_tile_minmax<<<BD_ * SPLIT_, 256, 0, stream>>>(cArr, sArr, bdpart);
    k_bd_fin     <<<1, 128, 0, stream>>>(bdpart, bdmm);
    k_resize     <<<NPIX / 256, 256, 0, stream>>>(cArr, sArr, bdmm, rArr, part);
    k_minmax_fin <<<1, 256, 0, stream>>>(part, NPIX / 256, fmm);
    k_writeout   <<<(NOUT + 255) / 256, 256, 0, stream>>>(rArr, fmm, out, NOUT);
}